// SSIM_38397007626583
// MI455X (gfx1250) — compile-verified
//
#include <hip/hip_runtime.h>
#include <hip/hip_bf16.h>

// ---------------------------------------------------------------------------
// SSIM via separable 11x11 Gaussian depthwise conv, mapped onto
// V_WMMA_F32_16X16X4_F32 (fp32 WMMA, so bit-compatible precision class with
// the fp32 reference). Each wave32 produces one 16x16 output tile.
//
// Conv-as-matmul:  Out(16xN) = T(16x28) * X(28xN), T[m][i] = g[i-m] (Toeplitz)
// K=28 padded -> 7 chained K=4 WMMAs. Vertical pass then horizontal pass.
// ---------------------------------------------------------------------------

typedef __attribute__((ext_vector_type(2))) float v2f;
typedef __attribute__((ext_vector_type(8))) float v8f;

#define IMG_H   512
#define IMG_W   512
#define N_IMGS  48            // B*C = 16*3
#define TILES_PER_IMG 1024    // 32*32 tiles of 16x16
#define TOTAL_TILES (N_IMGS * TILES_PER_IMG)      // 49152
#define WAVES_PER_BLOCK 4
#define BLOCK_THREADS (WAVES_PER_BLOCK * 32)
#define NUM_BLOCKS (TOTAL_TILES / WAVES_PER_BLOCK) // 12288

#define PS 32                 // padded plane row stride (floats)
#define PLANE_ELEMS (28 * PS) // 896
#define VT_STRIDE 36          // conflict-free for rows and D-store
#define VT_ELEMS (16 * VT_STRIDE) // 576
#define WAVE_SLOT (2 * PLANE_ELEMS + VT_ELEMS)    // 2368 floats / wave

static __device__ __forceinline__ v8f wmma4(v2f a, v2f b, v8f c) {
  // D = A(16x4,f32) * B(4x16,f32) + C(16x16,f32)
  return __builtin_amdgcn_wmma_f32_16x16x4_f32(
      /*neg_a=*/false, a, /*neg_b=*/false, b,
      /*c_mod=*/(short)0, c, /*reuse_a=*/false, /*reuse_b=*/false);
}

// One quantity (Q selects x, y, x*x, y*y, x*y): vertical conv (2 chunks),
// round-trip through vt, horizontal conv.  Returns 16x16 D-layout result.
template <int Q>
static __device__ __forceinline__ v8f conv_q(const float* xs, const float* ys,
                                             float* vt, const float coef[14],
                                             int idx, int h) {
#pragma unroll
  for (int cb = 0; cb < 2; ++cb) {
    v8f acc = {};
#pragma unroll
    for (int s = 0; s < 7; ++s) {
      const int K0 = 4 * s + 2 * h;            // element K for v=0
      const int i0 = K0 * PS + cb * 16 + idx;  // LDS index (row K0, this col)
      float b0, b1;
      if (Q == 0) { b0 = xs[i0];           b1 = xs[i0 + PS]; }
      else if (Q == 1) { b0 = ys[i0];      b1 = ys[i0 + PS]; }
      else if (Q == 2) { float a0 = xs[i0], a1 = xs[i0 + PS]; b0 = a0 * a0; b1 = a1 * a1; }
      else if (Q == 3) { float a0 = ys[i0], a1 = ys[i0 + PS]; b0 = a0 * a0; b1 = a1 * a1; }
      else { b0 = xs[i0] * ys[i0]; b1 = xs[i0 + PS] * ys[i0 + PS]; }
      v2f a; a.x = coef[2 * s]; a.y = coef[2 * s + 1];
      v2f b; b.x = b0; b.y = b1;
      acc = wmma4(a, b, acc);
    }
    // D layout: lane holds (M = v + 8h, N = idx); store to vt[M][cb*16+idx]
#pragma unroll
    for (int v = 0; v < 8; ++v)
      vt[(v + 8 * h) * VT_STRIDE + cb * 16 + idx] = acc[v];
  }
  // within-wave DS ordering is architectural, but fence explicitly
  asm volatile("s_wait_dscnt 0" ::: "memory");

  v8f acc2 = {};
#pragma unroll
  for (int s = 0; s < 7; ++s) {
    const int K0 = 4 * s + 2 * h;
    v2f a; a.x = vt[idx * VT_STRIDE + K0]; a.y = vt[idx * VT_STRIDE + K0 + 1];
    v2f b; b.x = coef[2 * s]; b.y = coef[2 * s + 1];
    acc2 = wmma4(a, b, acc2);
  }
  asm volatile("s_wait_dscnt 0" ::: "memory");
  return acc2;
}

__global__ __launch_bounds__(BLOCK_THREADS) void ssim_tile_kernel(
    const float* __restrict__ img1, const float* __restrict__ img2,
    const float* __restrict__ win, float* __restrict__ partial) {
  __shared__ float sm[WAVES_PER_BLOCK * WAVE_SLOT];
  __shared__ float g_sh[16];
  __shared__ float wsum[WAVES_PER_BLOCK];

  const int tid  = threadIdx.x;
  const int wave = tid >> 5;
  const int lane = tid & 31;
  const int idx  = lane & 15;   // M (A) / N (B,D) index
  const int h    = lane >> 4;   // half-wave selector

  // Derive the separable 1-D gaussian from the reference's 2-D window:
  // w2d[i][j] = g[i]*g[j]  ->  g[k] = w2d[5][k] / sqrt(w2d[5][5])
  if (tid < 11) {
    const float g5 = sqrtf(win[5 * 11 + 5]);
    g_sh[tid] = win[5 * 11 + tid] / g5;
  }
  __syncthreads();

  const int tile = blockIdx.x * WAVES_PER_BLOCK + wave;
  const int img  = tile >> 10;          // /1024
  const int t    = tile & 1023;
  const int tr   = t >> 5;
  const int tc   = t & 31;
  const long base = (long)img * (IMG_H * IMG_W);
  const int row0 = tr * 16 - 5;
  const int col0 = tc * 16 - 5;

  float* xs = sm + wave * WAVE_SLOT;
  float* ys = xs + PLANE_ELEMS;
  float* vt = ys + PLANE_ELEMS;

  // Load 28x32 padded tiles with replicate-edge clamping (finite everywhere;
  // Toeplitz zeros kill the overreach columns/rows mathematically).
  for (int i = lane; i < PLANE_ELEMS; i += 32) {
    const int r = i >> 5, c = i & 31;
    int gr = row0 + r; gr = gr < 0 ? 0 : (gr > IMG_H - 1 ? IMG_H - 1 : gr);
    int gc = col0 + c; gc = gc < 0 ? 0 : (gc > IMG_W - 1 ? IMG_W - 1 : gc);
    const long off = base + (long)gr * IMG_W + gc;
    xs[i] = img1[off];
    ys[i] = img2[off];
  }

  // Per-lane Toeplitz coefficients: coef[2s+v] = g[(4s+2h+v) - idx] or 0.
  float coef[14];
#pragma unroll
  for (int j = 0; j < 14; ++j) {
    const int s = j >> 1, v = j & 1;
    const int K = 4 * s + 2 * h + v;
    const int d = K - idx;
    coef[j] = ((unsigned)d <= 10u) ? g_sh[d] : 0.0f;
  }
  asm volatile("s_wait_dscnt 0" ::: "memory");

  const v8f mu1 = conv_q<0>(xs, ys, vt, coef, idx, h);
  const v8f mu2 = conv_q<1>(xs, ys, vt, coef, idx, h);
  const v8f e11 = conv_q<2>(xs, ys, vt, coef, idx, h);
  const v8f e22 = conv_q<3>(xs, ys, vt, coef, idx, h);
  const v8f e12 = conv_q<4>(xs, ys, vt, coef, idx, h);

  const float C1 = 0.01f * 0.01f;
  const float C2 = 0.03f * 0.03f;
  float lsum = 0.0f;
#pragma unroll
  for (int v = 0; v < 8; ++v) {
    const float m1 = mu1[v], m2 = mu2[v];
    const float m1s = m1 * m1, m2s = m2 * m2, m12 = m1 * m2;
    const float s1 = e11[v] - m1s;
    const float s2 = e22[v] - m2s;
    const float s12 = e12[v] - m12;
    const float num = (2.0f * m12 + C1) * (2.0f * s12 + C2);
    const float den = (m1s + m2s + C1) * (s1 + s2 + C2);
    lsum += num / den;
  }

  // wave32 reduction (deterministic)
#pragma unroll
  for (int off = 16; off > 0; off >>= 1)
    lsum += __shfl_down(lsum, off, 32);
  if (lane == 0) wsum[wave] = lsum;
  __syncthreads();
  if (tid == 0)
    partial[blockIdx.x] = wsum[0] + wsum[1] + wsum[2] + wsum[3];
}

__global__ __launch_bounds__(256) void ssim_reduce_kernel(
    const float* __restrict__ partial, float* __restrict__ out, int n,
    float scale) {
  __shared__ float sh[256];
  const int tid = threadIdx.x;
  float s = 0.0f;
  for (int i = tid; i < n; i += 256) s += partial[i];
  sh[tid] = s;
  __syncthreads();
  for (int o = 128; o > 0; o >>= 1) {
    if (tid < o) sh[tid] += sh[tid + o];
    __syncthreads();
  }
  if (tid == 0) out[0] = sh[0] * scale;
}

extern "C" void kernel_launch(void* const* d_in, const int* in_sizes, int n_in,
                              void* d_out, int out_size, void* d_ws,
                              size_t ws_size, hipStream_t stream) {
  const float* img1 = (const float*)d_in[0];
  const float* img2 = (const float*)d_in[1];
  const float* win  = (const float*)d_in[2];
  float* out = (float*)d_out;
  float* partial = (float*)d_ws;  // NUM_BLOCKS floats = 48 KiB

  ssim_tile_kernel<<<NUM_BLOCKS, BLOCK_THREADS, 0, stream>>>(img1, img2, win,
                                                             partial);
  const float scale = 1.0f / (float)(N_IMGS * IMG_H * IMG_W);  // 1/12582912
  ssim_reduce_kernel<<<1, 256, 0, stream>>>(partial, out, NUM_BLOCKS, scale);
}